// LGE_Convolution_84550726189151
// MI455X (gfx1250) — compile-verified
//
#include <hip/hip_runtime.h>

// LGE convolution for gfx1250 (MI455X): WMMA weight contraction + async
// global->LDS staged features (double-buffered) + VALU complex transport.
//   H_{k,l} = sum_i w[i,k,l] F_i     : V_WMMA_F32_16X16X4_F32 (4 chained K=4)
//   out    += A_l H A_l^dag          : per-lane pk-f32 complex math, streamed
//                                      per H-column (min register pressure)
//   F tile staging                   : GLOBAL_LOAD_ASYNC_TO_LDS_B64/B32,
//                                      tracked by ASYNCcnt, double-buffered.

typedef __attribute__((ext_vector_type(2))) float v2f;
typedef __attribute__((ext_vector_type(8))) float v8f;

#define V_SITES 131072   // 16*16*16*32
#define NE 9             // 3x3 matrix elements
#define NL 9             // paths: identity + 8 hops
#define NK 8             // n_out
#define CH_BYTES 1152    // per-channel staged block: 16 sites * 9 elems * 8B
#define FCH_BYTES 9437184ull  // bytes per F channel = V_SITES*NE*8

typedef v2f c2;          // complex as (re, im): lowers to v_pk_fma_f32
__device__ __forceinline__ c2 cmac(c2 s, c2 a, c2 b) {           // s += a*b
    s += (c2){a.x, a.x} * b;
    s += (c2){-a.y, a.y} * (c2){b.y, b.x};
    return s;
}
__device__ __forceinline__ c2 cmacj(c2 s, c2 a, c2 b) {          // s += a*conj(b)
    s += (c2){a.x, a.y} * (c2){b.x, b.x};
    s += (c2){a.y, -a.x} * (c2){b.y, b.y};
    return s;
}
__device__ __forceinline__ c2 cjmac(c2 s, c2 a, c2 b) {          // s += conj(a)*b
    s += (c2){a.x, a.x} * b;
    s += (c2){a.y, -a.y} * (c2){b.y, b.x};
    return s;
}

__device__ __forceinline__ int SITE(int x, int y, int z, int t) {
    return ((x * 16 + y) * 16 + z) * 32 + t;
}

// Async-stage one path block (8 channels x 1152B contiguous) into LDS.
// Global: GVS mode  addr = SADDR(channel base) + VADDR(sblock*72 + lane*sz) + imm
// LDS:    dsaddr    = VDST(lbuf + i*1152 + lane*sz) + imm   (same imm, by ISA)
__device__ __forceinline__ void stageF(const float2* __restrict__ F2,
                                       int sblock, unsigned lbuf, int lane) {
    const unsigned vb64 = (unsigned)sblock * 72u + (unsigned)lane * 8u;
    const unsigned vb32 = (unsigned)sblock * 72u + (unsigned)lane * 4u;
    #pragma unroll
    for (int i = 0; i < 8; ++i) {
        const unsigned long long sbase =
            (unsigned long long)F2 + (unsigned long long)i * FCH_BYTES;
        const unsigned l64 = lbuf + (unsigned)i * CH_BYTES + (unsigned)lane * 8u;
        const unsigned l32 = lbuf + (unsigned)i * CH_BYTES + (unsigned)lane * 4u;
        asm volatile("global_load_async_to_lds_b64 %0, %1, %2 offset:0"
                     :: "v"(l64), "v"(vb64), "s"(sbase) : "memory");
        asm volatile("global_load_async_to_lds_b64 %0, %1, %2 offset:256"
                     :: "v"(l64), "v"(vb64), "s"(sbase) : "memory");
        asm volatile("global_load_async_to_lds_b64 %0, %1, %2 offset:512"
                     :: "v"(l64), "v"(vb64), "s"(sbase) : "memory");
        asm volatile("global_load_async_to_lds_b64 %0, %1, %2 offset:768"
                     :: "v"(l64), "v"(vb64), "s"(sbase) : "memory");
        asm volatile("global_load_async_to_lds_b32 %0, %1, %2 offset:1024"
                     :: "v"(l32), "v"(vb32), "s"(sbase) : "memory");
    }
}

__global__ void __launch_bounds__(256)
lge_conv_wmma(const float2* __restrict__ U2,   // [4][V][9] complex
              const float2* __restrict__ F2,   // [8][V][9] complex
              const float2* __restrict__ W2,   // [8][8][9] complex
              float2* __restrict__ O2)         // [8][V][9] complex
{
    // Expanded real weight matrices (16x16 per path), row stride 18 floats.
    __shared__ float Aexp[NL][16][18];
    // Double-buffered staged F blocks: [wave][buf][channel][site][elem].
    __shared__ __align__(16) float2 Fstage[8][2][8][16][NE];

    const int tid = threadIdx.x;
    for (int idx = tid; idx < NL * 256; idx += 256) {
        const int l  = idx >> 8;
        const int r  = idx & 255;
        const int m  = r >> 4;        // 2k + c
        const int kk = r & 15;        // 2i + d
        const int k = m >> 1, c = m & 1, i = kk >> 1, d = kk & 1;
        const float2 w = W2[(i * NK + k) * NL + l];
        Aexp[l][m][kk] = (c == 0) ? (d == 0 ? w.x : -w.y)
                                  : (d == 0 ? w.y :  w.x);
    }
    __syncthreads();

    const int lane = tid & 31;
    const int wave = tid >> 5;
    const int tile = blockIdx.x * 8 + wave;    // 8192 tiles of 16 t-sites
    const int n    = lane & 15;                // site-in-tile == WMMA N column
    const int hi   = lane >> 4;                // lane half (K rows / k 4..7)

    const int th = tile & 1;
    const int z  = (tile >> 1) & 15;
    const int y  = (tile >> 5) & 15;
    const int x  = (tile >> 9) & 15;
    const int t0 = th * 16;
    const int t  = t0 + n;

    // per-lane element offsets (site*9) for own site + 8 shifted sites (U, l7/8 F)
    int soff[NL];
    soff[0] = SITE(x, y, z, t) * NE;
    soff[1] = SITE((x + 1) & 15, y, z, t) * NE;
    soff[2] = SITE((x + 15) & 15, y, z, t) * NE;
    soff[3] = SITE(x, (y + 1) & 15, z, t) * NE;
    soff[4] = SITE(x, (y + 15) & 15, z, t) * NE;
    soff[5] = SITE(x, y, (z + 1) & 15, t) * NE;
    soff[6] = SITE(x, y, (z + 15) & 15, t) * NE;
    soff[7] = SITE(x, y, z, (t + 1) & 31) * NE;
    soff[8] = SITE(x, y, z, (t + 31) & 31) * NE;

    // wave-uniform block base sites for the 7 contiguous (non-t-shift) paths
    int sblock[7];
    sblock[0] = SITE(x, y, z, t0);
    sblock[1] = SITE((x + 1) & 15, y, z, t0);
    sblock[2] = SITE((x + 15) & 15, y, z, t0);
    sblock[3] = SITE(x, (y + 1) & 15, z, t0);
    sblock[4] = SITE(x, (y + 15) & 15, z, t0);
    sblock[5] = SITE(x, y, (z + 1) & 15, t0);
    sblock[6] = SITE(x, y, (z + 15) & 15, t0);

    const unsigned lbuf0 = (unsigned)(unsigned long long)&Fstage[wave][0][0][0][0];
    const unsigned lbuf1 = (unsigned)(unsigned long long)&Fstage[wave][1][0][0][0];

    // Persistent accumulators: scalar complex [q = local out channel][e].
    c2 oacc[4][NE];

    stageF(F2, sblock[0], lbuf0, lane);   // prefetch path 0

    #pragma unroll
    for (int l = 0; l < NL; ++l) {
        // prefetch next staged path into the other buffer
        if (l + 1 <= 6)
            stageF(F2, sblock[l + 1], ((l + 1) & 1) ? lbuf1 : lbuf0, lane);
        // ensure this path's buffer is complete (async ops finish in order)
        if (l <= 5)      asm volatile("s_wait_asynccnt 0x28" ::: "memory");
        else if (l == 6) asm volatile("s_wait_asynccnt 0x0"  ::: "memory");

        const bool staged = (l <= 6);
        const int  p      = l & 1;

        v2f afr[4];
        #pragma unroll
        for (int j = 0; j < 4; ++j)
            afr[j] = *(const v2f*)&Aexp[l][n][4 * j + hi * 2];

        const int  mu  = (l - 1) >> 1;
        const bool fwd = ((l - 1) & 1) == 0;
        c2 Um[3][3];
        if (l > 0) {
            const int uoff = fwd ? soff[0] : soff[l];
            const float2* __restrict__ Umu = U2 + (size_t)mu * (V_SITES * NE);
            #pragma unroll
            for (int e = 0; e < NE; ++e) {
                const float2 u = Umu[uoff + e];
                Um[e / 3][e % 3] = (c2){u.x, u.y};
            }
        }

        // Stream H by columns b (3 elements each): 12 WMMAs per column.
        #pragma unroll
        for (int b = 0; b < 3; ++b) {
            v8f col[3];
            #pragma unroll
            for (int c = 0; c < 3; ++c) {
                const int e = 3 * c + b;
                v8f acc = (v8f){};
                #pragma unroll
                for (int j = 0; j < 4; ++j) {
                    const int i = 2 * j + hi;
                    float2 f;
                    if (staged) f = Fstage[wave][p][i][n][e];                   // ds_load_b64
                    else        f = (F2 + (size_t)i * (V_SITES * NE))[soff[l] + e];
                    v2f bfr; bfr.x = f.x; bfr.y = f.y;
                    acc = __builtin_amdgcn_wmma_f32_16x16x4_f32(
                        false, afr[j], false, bfr, (short)0, acc, false, false);
                }
                col[c] = acc;
            }

            if (l == 0) {           // identity path: initialize oacc
                #pragma unroll
                for (int c = 0; c < 3; ++c)
                    #pragma unroll
                    for (int q = 0; q < 4; ++q)
                        oacc[q][3 * c + b] = (c2){col[c][2 * q], col[c][2 * q + 1]};
            } else {
                #pragma unroll
                for (int q = 0; q < 4; ++q) {
                    c2 h[3];
                    #pragma unroll
                    for (int c = 0; c < 3; ++c)
                        h[c] = (c2){col[c][2 * q], col[c][2 * q + 1]};
                    // stage 1: P[a][b] = sum_c U'[a][c] H[c][b]
                    c2 Pcol[3];
                    #pragma unroll
                    for (int a = 0; a < 3; ++a) {
                        c2 s = (c2){};
                        if (fwd) { s = cmac(s, Um[a][0], h[0]);
                                   s = cmac(s, Um[a][1], h[1]);
                                   s = cmac(s, Um[a][2], h[2]); }
                        else     { s = cjmac(s, Um[0][a], h[0]);
                                   s = cjmac(s, Um[1][a], h[1]);
                                   s = cjmac(s, Um[2][a], h[2]); }
                        Pcol[a] = s;
                    }
                    // stage 2 scatter: R[a][b2] += P[a][b] * U''[b][b2]
                    #pragma unroll
                    for (int a = 0; a < 3; ++a)
                        #pragma unroll
                        for (int b2 = 0; b2 < 3; ++b2) {
                            if (fwd)
                                oacc[q][3 * a + b2] = cmacj(oacc[q][3 * a + b2], Pcol[a], Um[b2][b]);
                            else
                                oacc[q][3 * a + b2] = cmac (oacc[q][3 * a + b2], Pcol[a], Um[b][b2]);
                        }
                }
            }
        }
    }

    // store (non-temporal: output is write-only; keep L2 for F/U reuse)
    #pragma unroll
    for (int q = 0; q < 4; ++q) {
        const int k = q + hi * 4;
        v2f* __restrict__ Ok = (v2f*)(O2 + (size_t)k * (V_SITES * NE) + soff[0]);
        #pragma unroll
        for (int e = 0; e < NE; ++e)
            __builtin_nontemporal_store(oacc[q][e], &Ok[e]);
    }
}

extern "C" void kernel_launch(void* const* d_in, const int* in_sizes, int n_in,
                              void* d_out, int out_size, void* d_ws, size_t ws_size,
                              hipStream_t stream) {
    const float2* U2 = (const float2*)d_in[0];   // complex64 -> interleaved float2
    const float2* F2 = (const float2*)d_in[1];
    const float2* W2 = (const float2*)d_in[2];
    float2* O2 = (float2*)d_out;

    dim3 grid(1024), block(256);  // 8192 wave-tiles x 16 sites = 131072 sites
    hipLaunchKernelGGL(lge_conv_wmma, grid, block, 0, stream, U2, F2, W2, O2);
}